// CompositeBezierCurve_25709674234606
// MI455X (gfx1250) — compile-verified
//
#include <hip/hip_runtime.h>
#include <hip/hip_bf16.h>

// Composite cubic Bezier evaluation, MI455X (gfx1250).
// Memory/latency-bound gather workload:
//  - uniform LUT (d_ws) turns the 17-step binary search into ~1 L2 load
//  - b128 control-point gathers (48B rows, 16B aligned)
//  - global_prefetch_b8 overlaps the cp gather with the knot scan
//  - non-temporal loads/stores keep the streamed 160MB from evicting the
//    ~7MB hot gather set (LUT+cp+knots) out of the 192MB L2
//  - LDS coarse-level fallback search if ws is too small for the LUT

#define LUT_LOG2 19
#define LUT_M (1 << LUT_LOG2)   // cell width <= x_end/2^19 < 0.29 < min(dx)=0.5

// ---------------------------------------------------------------------------
// Shared per-point evaluation (after segment index j and k0/k1 are known).
// ---------------------------------------------------------------------------
__device__ __forceinline__ void bez_eval_point(
    const float* __restrict__ cp,
    int j, float r, float k0, float k1, int i,
    float* __restrict__ out_vals, float* __restrict__ out_idx)
{
    float s = (r - k0) / (k1 - k0);

    // Bernstein basis, same op order as the reference's cumprod construction.
    float t2 = s * s;
    float t3 = t2 * s;
    float u  = 1.0f - s;
    float u2 = u * u;
    float u3 = u2 * u;
    float b0 = u3;
    float b1 = (3.0f * s) * u2;
    float b2 = (3.0f * t2) * u;
    float b3 = t3;

    // 12 contiguous floats, 16B aligned (stride 48B) -> 3x global_load_b128.
    const float4* cp4 = reinterpret_cast<const float4*>(cp) + (size_t)j * 3;
    float4 a = cp4[0];  // P0x P0y P0z P1x
    float4 b = cp4[1];  // P1y P1z P2x P2y
    float4 c = cp4[2];  // P2z P3x P3y P3z

    float vx = fmaf(b0, a.x, fmaf(b1, a.w, fmaf(b2, b.z, b3 * c.y)));
    float vy = fmaf(b0, a.y, fmaf(b1, b.x, fmaf(b2, b.w, b3 * c.z)));
    float vz = fmaf(b0, a.z, fmaf(b1, b.y, fmaf(b2, c.x, b3 * c.w)));

    // Write-once outputs: TH_STORE_NT so the 128MB stream does not rinse
    // the L2-resident gather working set.
    size_t o = (size_t)i * 3;
    __builtin_nontemporal_store(vx, out_vals + o + 0);
    __builtin_nontemporal_store(vy, out_vals + o + 1);
    __builtin_nontemporal_store(vz, out_vals + o + 2);
    __builtin_nontemporal_store((float)j, out_idx + i);  // idx <= 1e5 exact in f32
}

__device__ __forceinline__ float mod_pos(float x, float xe)
{
    // Matches jnp.mod (= lax.rem + sign fix) for x in [-0.5*xe, 1.5*xe).
    float r = fmodf(x, xe);
    return (r < 0.0f) ? r + xe : r;
}

// ---------------------------------------------------------------------------
// Kernel 1: build the uniform LUT.  lut[g] = searchsorted(xstart, g*w,'right')-1
// clipped to [0, nseg-1].  M+1 entries, each via a log2(nseg) binary search.
// Cheap (524K threads) and the LUT stays hot in the 192MB L2.
// ---------------------------------------------------------------------------
__global__ __launch_bounds__(256) void bez_build_lut(
    const float* __restrict__ knots, int nseg, int* __restrict__ lut)
{
    int t = blockIdx.x * blockDim.x + threadIdx.x;
    if (t > LUT_M) return;
    float xe = knots[nseg];
    float xg = (float)t * (xe / (float)LUT_M);
    int lo = 0, hi = nseg;
    while (lo < hi) {
        int mid = (lo + hi) >> 1;
        if (knots[mid] <= xg) lo = mid + 1; else hi = mid;
    }
    int idx = lo - 1;
    idx = (idx < 0) ? 0 : ((idx > nseg - 1) ? nseg - 1 : idx);
    lut[t] = idx;
}

// ---------------------------------------------------------------------------
// Kernel 2: main evaluation with O(1) LUT lookup + robust +-1 scan that
// carries k0/k1 in registers (shortest possible dependent-load chain).
// ---------------------------------------------------------------------------
__global__ __launch_bounds__(256) void bez_eval_lut(
    const float* __restrict__ knots, const float* __restrict__ cp,
    const float* __restrict__ xev, const int* __restrict__ lut,
    int nseg, int npts,
    float* __restrict__ out_vals, float* __restrict__ out_idx)
{
    int i = blockIdx.x * blockDim.x + threadIdx.x;
    if (i >= npts) return;

    float xe = knots[nseg];                         // uniform -> scalar load
    float x  = __builtin_nontemporal_load(xev + i); // read-once stream: TH_LOAD_NT
    float r  = mod_pos(x, xe);

    float scale = (float)LUT_M / xe;
    int g = (int)(r * scale);
    g = (g < 0) ? 0 : ((g > LUT_M - 1) ? LUT_M - 1 : g);

    int j = lut[g];                                 // hint; exact up to +-1 step

    // gfx1250: warm the control-point row while we finish the knot scan.
    __builtin_prefetch(cp + (size_t)j * 12, 0, 3);

    // searchsorted(side='right')-1 semantics via '<='; LUT is only a hint so
    // the result is independent of its float quantization.
    float k0 = knots[j];
    while (j > 0 && k0 > r) { --j; k0 = knots[j]; }
    float k1 = knots[j + 1];
    while (j < nseg - 1 && k1 <= r) { ++j; k0 = k1; k1 = knots[j + 1]; }

    bez_eval_point(cp, j, r, k0, k1, i, out_vals, out_idx);
}

// ---------------------------------------------------------------------------
// Fallback (ws too small): LDS-staged coarse knot level (<=512 entries)
// + short global binary refine.  Halves the global dependent-load chain.
// ---------------------------------------------------------------------------
__global__ __launch_bounds__(256) void bez_eval_bsearch(
    const float* __restrict__ knots, const float* __restrict__ cp,
    const float* __restrict__ xev, int nseg, int npts,
    float* __restrict__ out_vals, float* __restrict__ out_idx)
{
    __shared__ float coarse[512];
    int cstride = (nseg + 511) / 512;            // knots per coarse cell
    int C       = (nseg + cstride - 1) / cstride;
    for (int t = threadIdx.x; t < C; t += blockDim.x)
        coarse[t] = knots[(size_t)t * cstride];
    __syncthreads();

    int i = blockIdx.x * blockDim.x + threadIdx.x;
    if (i >= npts) return;

    float xe = knots[nseg];
    float x  = __builtin_nontemporal_load(xev + i);
    float r  = mod_pos(x, xe);

    // Coarse search in LDS: largest c with coarse[c] <= r.
    int lo = 0, hi = C;
    while (lo < hi) {
        int mid = (lo + hi) >> 1;
        if (coarse[mid] <= r) lo = mid + 1; else hi = mid;
    }
    int c = lo - 1;
    c = (c < 0) ? 0 : c;

    // Fine search in global within one coarse cell.
    int lo2 = c * cstride;
    int hi2 = lo2 + cstride;
    hi2 = (hi2 > nseg) ? nseg : hi2;
    while (lo2 < hi2) {
        int mid = (lo2 + hi2) >> 1;
        if (knots[mid] <= r) lo2 = mid + 1; else hi2 = mid;
    }
    int j = lo2 - 1;
    j = (j < 0) ? 0 : ((j > nseg - 1) ? nseg - 1 : j);

    float k0 = knots[j];
    float k1 = knots[j + 1];
    bez_eval_point(cp, j, r, k0, k1, i, out_vals, out_idx);
}

// ---------------------------------------------------------------------------
extern "C" void kernel_launch(void* const* d_in, const int* in_sizes, int n_in,
                              void* d_out, int out_size, void* d_ws, size_t ws_size,
                              hipStream_t stream)
{
    (void)n_in; (void)out_size;
    const float* knots = (const float*)d_in[0];   // (N_SEG+1,)
    const float* cp    = (const float*)d_in[1];   // (N_SEG, 4, 3)
    const float* xev   = (const float*)d_in[2];   // (BATCH*NPTS,)

    int nseg = in_sizes[0] - 1;
    int npts = in_sizes[2];

    float* out_vals = (float*)d_out;                      // npts*3 floats
    float* out_idx  = out_vals + (size_t)npts * 3;        // npts floats

    const int block   = 256;                              // 8 wave32 waves
    const int eblocks = (npts + block - 1) / block;

    size_t lut_bytes = (size_t)(LUT_M + 1) * sizeof(int);
    if (ws_size >= lut_bytes) {
        int* lut = (int*)d_ws;
        int lblocks = (LUT_M + 1 + block - 1) / block;
        bez_build_lut<<<lblocks, block, 0, stream>>>(knots, nseg, lut);
        bez_eval_lut<<<eblocks, block, 0, stream>>>(knots, cp, xev, lut,
                                                    nseg, npts, out_vals, out_idx);
    } else {
        bez_eval_bsearch<<<eblocks, block, 0, stream>>>(knots, cp, xev,
                                                        nseg, npts, out_vals, out_idx);
    }
}